// HeatSolver_36825049596438
// MI455X (gfx1250) — compile-verified
//
#include <hip/hip_runtime.h>
#include <stdint.h>
#include <math.h>

// ---------------------------------------------------------------------------
// 2D heat equation, 256x256 grid, 512 explicit steps, full history output.
// Strategy (MI455X / gfx1250):
//  * whole u grid (256 KB) lives in one WGP's 320 KB LDS -> single persistent
//    workgroup runs all 512 steps with only workgroup barriers (the problem is
//    sync-latency bound: ~0.4 GFLOP total, output traffic ~134 MB ≈ 6 us HBM).
//  * per-step 256 KB history snapshot streamed out with the Tensor Data Mover
//    (tensor_store_from_lds, TENSORcnt), overlapped with the next sub-phase's
//    compute; s_wait_tensorcnt(12) pipelines exactly one full step of TDM ops.
//  * WMMA is deliberately NOT used: a 5-point stencil with per-cell
//    coefficients has no GEMM structure; matrix ops would add work.
// ---------------------------------------------------------------------------

#define MM   256
#define NB   1020          /* 4*(MM-1) boundary cells            */
#define SUB  16            /* rows per sub-phase                  */
#define RPB  64            /* rows per band (4 bands x 256 cols)  */

typedef unsigned int v4u __attribute__((ext_vector_type(4)));
typedef int          v8i __attribute__((ext_vector_type(8)));
typedef int          v4i __attribute__((ext_vector_type(4)));

#if defined(__has_builtin)
#if __has_builtin(__builtin_amdgcn_tensor_store_from_lds) && \
    __has_builtin(__builtin_amdgcn_s_wait_tensorcnt)
#define HAVE_TDM 1
#endif
#endif

#ifdef HAVE_TDM
#define WAIT_TENSORCNT(n) __builtin_amdgcn_s_wait_tensorcnt(n)
#else
#define WAIT_TENSORCNT(n)
#endif

#ifdef HAVE_TDM
// Contiguous LDS -> global copy of `nelem` fp32 elements via the TDM.
// D# group0: count=1(valid), lds byte addr, 57-bit global addr, type=2.
// D# group1: data_size=4B; tensor_dim0=tile_dim0=nelem; tensor_dim1=tile_dim1=1;
//            tensor_dim0_stride=nelem.  Groups 2/3 (and the trailing v8i of the
//            6-arg clang-23 builtin form) are zero: tensor is <= 2-D.
__device__ __forceinline__ void tdm_store_contig(const void* lds_src,
                                                 const float* gdst,
                                                 unsigned nelem) {
    unsigned long long ga = (unsigned long long)(uintptr_t)gdst;
    unsigned lds_byte = (unsigned)(uintptr_t)lds_src;  // low 32b = LDS offset
    v4u g0 = { 1u,
               lds_byte,
               (unsigned)(ga & 0xffffffffull),
               (unsigned)((ga >> 32) & 0x1ffffffull) | (2u << 30) };
    v8i g1 = { (int)(2u << 16),            /* data_size = 4 bytes            */
               (int)(nelem << 16),         /* tensor_dim0[15:0]  @ bits63:48 */
               (int)((nelem >> 16) | (1u << 16)), /* dim0 hi | tensor_dim1=1 */
               (int)(nelem << 16),         /* tile_dim0 @ bits127:112        */
               1,                          /* tile_dim1 = 1                  */
               (int)nelem,                 /* tensor_dim0_stride low32       */
               0, 0 };
    v4i gz4 = {0, 0, 0, 0};
    v8i gz8 = {0, 0, 0, 0, 0, 0, 0, 0};
    __builtin_amdgcn_tensor_store_from_lds(g0, g1, gz4, gz4, gz8, 0);
}
#endif

// ---------------------------------------------------------------------------
// Setup: fold tau/h^2 into harmonic interface conductivities, fold tau into
// the spatial source factor, zero history slot 0.
// ws layout (floats): cE[65536] | cN[65536] | gsrc[65536]
// (cN[-1] aliases cE[255][255] == 0, making the j==0 south term branch-free.)
// ---------------------------------------------------------------------------
__global__ void heat_setup(const float* __restrict__ sigma,
                           const float* __restrict__ Tp,
                           const int* __restrict__ nsp,
                           float* __restrict__ cE, float* __restrict__ cN,
                           float* __restrict__ gs, float* __restrict__ out) {
    const int idx = blockIdx.x * blockDim.x + threadIdx.x;   // 0..65535
    const int i = idx >> 8, j = idx & (MM - 1);
    const int n = *nsp;
    const float tau = *Tp / (float)n;
    const float h = 1.0f / (float)MM;
    const float coef = tau / (h * h);
    const float sc = sigma[idx];
    float ce = 0.0f, cn = 0.0f;
    if (i < MM - 1) { float sb = sigma[idx + MM]; ce = coef * 2.0f * sc * sb / (sc + sb); }
    if (j < MM - 1) { float sb = sigma[idx + 1];  cn = coef * 2.0f * sc * sb / (sc + sb); }
    cE[idx] = ce;
    cN[idx] = cn;
    const float x = (i + 0.5f) * h - 0.5f;
    const float y = (j + 0.5f) * h - 0.5f;
    gs[idx] = tau * expf(-(x * x + y * y) / 0.02f);
    // u_history[0] = 0, u_b_history[0] = 0
    float* uh0 = out + MM * MM + (size_t)(n + 1) * NB;
    uh0[idx] = 0.0f;
    if (idx < NB) out[MM * MM + idx] = 0.0f;
}

// ---------------------------------------------------------------------------
// Persistent stepper: one workgroup, 1024 threads (32 waves, one full WGP).
// Thread t owns column j = t&255 and the 64-row band (t>>8).  Each step runs
// as 4 sub-phases of 16 rows: compute into regs (old values except i+1 of a band's last row; the
// vertical predecessor and west coefficient roll through registers across
// sub-phase boundaries) -> wait/barrier -> write LDS + boundary history ->
// barrier -> thread 0 issues 4 TDM strip stores (16 per step, in-order).
// TENSORcnt<=12 before each write phase == strips written one full step ago
// have drained.
// ---------------------------------------------------------------------------
__global__ __launch_bounds__(1024, 1)
void heat_run(const float* __restrict__ cE, const float* __restrict__ cN,
              const float* __restrict__ gs,
              const float* __restrict__ Tp, const int* __restrict__ nsp,
              float* __restrict__ out) {
    __shared__ float sm[(MM + 2) * MM];   // pad row before + after: 264 KB LDS
    float* u = sm + MM;
    const int tid  = threadIdx.x;
    const int j    = tid & (MM - 1);
    const int band = tid >> 8;
    const int i0   = band * RPB;
    const int n    = *nsp;
    const float tau = *Tp / (float)n;
    float* ubase = out + MM * MM;                       // boundary history
    float* uhist = ubase + (size_t)(n + 1) * NB;        // full history

    for (int x = tid; x < (MM + 2) * MM; x += blockDim.x) sm[x] = 0.0f;
    __syncthreads();

    for (int k = 0; k < n; ++k) {
        const float s = 1.0f + sinf(10.0f * (float)k * tau);
        float u_im1, cW;
        if (band == 0) { u_im1 = 0.0f; cW = 0.0f; }            // zero-flux BC
        else { u_im1 = u[(i0 - 1) * MM + j]; cW = cE[(i0 - 1) * MM + j]; }
        float* ub = ubase + (size_t)(k + 1) * NB;
        float* uh = uhist + (size_t)(k + 1) * (MM * MM);

        for (int sb = 0; sb < RPB / SUB; ++sb) {
            const int rbase = i0 + sb * SUB;
            float un[SUB];
            #pragma unroll
            for (int c = 0; c < SUB; ++c) {
                const int idx = (rbase + c) * MM + j;
                const float uc  = u[idx];
                const float uip = u[idx + MM];   // i==255 -> pad row (x ce=0)
                const float ujm = u[idx - 1];    // j==0   -> neighbor (x cs=0)
                const float ujp = u[idx + 1];    // j==255 -> neighbor (x cn=0)
                const float ce  = cE[idx];
                const float cn  = cN[idx];
                const float cs  = cN[idx - 1];   // ==0 whenever j==0
                const float g   = gs[idx];
                un[c] = uc + ce * (uip - uc) - cW * (uc - u_im1)
                           + cn * (ujp - uc) - cs * (uc - ujm) + g * s;
                u_im1 = uc;
                cW = ce;
            }
            // Strips about to be overwritten were TDM'd 16 issues ago.
            WAIT_TENSORCNT(12);
            __syncthreads();
            #pragma unroll
            for (int c = 0; c < SUB; ++c) {
                const int i = rbase + c;
                const int idx = i * MM + j;
                const float v = un[c];
                u[idx] = v;
#ifndef HAVE_TDM
                uh[idx] = v;                     // fallback history store
#endif
                if (i == 0)            ub[j] = v;
                else if (i == MM - 1)  ub[(NB - MM) + j] = v;        // 764 + j
                else if (j == 0)       ub[MM + (i - 1) * 2] = v;
                else if (j == MM - 1)  ub[MM + (i - 1) * 2 + 1] = v;
            }
            __syncthreads();
#ifdef HAVE_TDM
            if (tid == 0) {
                #pragma unroll
                for (int b = 0; b < 4; ++b) {
                    const int r = b * RPB + sb * SUB;   // strip just written
                    tdm_store_contig(&u[r * MM], uh + r * MM, SUB * MM);
                }
            }
#endif
        }
    }
    __syncthreads();
    #pragma unroll 4
    for (int c = 0; c < RPB; ++c) {
        const int idx = (i0 + c) * MM + j;
        out[idx] = u[idx];                       // u_final
    }
    // s_endpgm's implicit wait-idle drains remaining TENSORcnt.
}

extern "C" void kernel_launch(void* const* d_in, const int* in_sizes, int n_in,
                              void* d_out, int out_size, void* d_ws, size_t ws_size,
                              hipStream_t stream) {
    const float* sigma = (const float*)d_in[0];
    const float* T     = (const float*)d_in[1];
    const int*   nst   = (const int*)d_in[2];
    float* out = (float*)d_out;
    float* cE = (float*)d_ws;            // 65536 floats
    float* cN = cE + MM * MM;            // 65536 floats (cN[-1] == cE[65535] == 0)
    float* gs = cN + MM * MM;            // 65536 floats
    heat_setup<<<(MM * MM) / 256, 256, 0, stream>>>(sigma, T, nst, cE, cN, gs, out);
    heat_run<<<1, 1024, 0, stream>>>(cE, cN, gs, T, nst, out);
}